// AvgFeatAGG2d_28982439313581
// MI455X (gfx1250) — compile-verified
//
#include <hip/hip_runtime.h>
#include <cstdint>

// 3x3 VALID sliding-window average over 4096 independent 64x64 f32 planes.
// Bandwidth-bound (130 MB traffic, ~142 MFLOP): optimized around the CDNA5
// async global->LDS DMA path + separable filtering in LDS.

#define TILE    64
#define OUTW    62
#define THREADS 256

__global__ __launch_bounds__(THREADS)
void avg3x3_async_kernel(const float* __restrict__ x, float* __restrict__ out) {
    __shared__ float tile[TILE * TILE];   // 16 KB: raw 64x64 input plane
    __shared__ float rsum[TILE * TILE];   // 16 KB: horizontal 3-tap sums (row stride 64)

    const int tid = threadIdx.x;
    const int img = blockIdx.x;                       // one (n,c) plane per block
    const float* __restrict__ src = x + (uint64_t)img * (TILE * TILE);

    // ---- Stage 1: async DMA global -> LDS (CDNA5 GLOBAL_LOAD_ASYNC_TO_LDS) ----
    // 4096 floats = 16 KB; 256 threads x 4 chunks x b128 (16 B/lane), fully
    // coalesced (lane stride 16 B -> 512 B per wave per issue).
    {
        #pragma unroll
        for (int j = 0; j < 4; ++j) {
            unsigned elem    = (unsigned)(j * THREADS + tid) * 4u;  // float index
            unsigned goff    = elem * 4u;                           // global byte offset (GVS vaddr)
            unsigned lds_off = (unsigned)(uintptr_t)(const void*)&tile[elem]; // low 32b = LDS byte addr
            asm volatile("global_load_async_to_lds_b128 %0, %1, %2"
                         :
                         : "v"(lds_off), "v"(goff), "s"(src)
                         : "memory");
        }
        // Wait for this wave's async copies to land in LDS (ASYNCcnt == 0).
        asm volatile("s_wait_asynccnt 0x0" ::: "memory");
    }
    __syncthreads();   // make all waves' DMA'd data visible workgroup-wide

    // ---- Stage 2: horizontal 3-tap sums: rsum[y][x] = t[y][x]+t[y][x+1]+t[y][x+2]
    // 64 rows x 62 cols; consecutive lanes hit consecutive LDS banks.
    for (int idx = tid; idx < TILE * OUTW; idx += THREADS) {
        int y    = idx / OUTW;
        int xo   = idx - y * OUTW;
        int base = y * TILE + xo;
        rsum[base] = tile[base] + tile[base + 1] + tile[base + 2];
    }
    __syncthreads();

    // ---- Stage 3: vertical 3-tap + scale; contiguous coalesced stores ----
    const float inv9 = 1.0f / 9.0f;
    float* __restrict__ dst = out + (uint64_t)img * (OUTW * OUTW);
    for (int idx = tid; idx < OUTW * OUTW; idx += THREADS) {
        int y    = idx / OUTW;
        int xo   = idx - y * OUTW;
        int base = y * TILE + xo;
        dst[idx] = (rsum[base] + rsum[base + TILE] + rsum[base + 2 * TILE]) * inv9;
    }
}

extern "C" void kernel_launch(void* const* d_in, const int* in_sizes, int n_in,
                              void* d_out, int out_size, void* d_ws, size_t ws_size,
                              hipStream_t stream) {
    (void)n_in; (void)out_size; (void)d_ws; (void)ws_size;
    const float* x = (const float*)d_in[0];
    float* out     = (float*)d_out;
    const int n_imgs = in_sizes[0] / (TILE * TILE);   // 16*256 = 4096 planes
    avg3x3_async_kernel<<<n_imgs, THREADS, 0, stream>>>(x, out);
}